// blobDiceLoss_2800318677066
// MI455X (gfx1250) — compile-verified
//
#include <hip/hip_runtime.h>

// Problem constants (reference: B=2, C=2, D=128, K=5 blobs -> labels 0..5)
#define NVOX   (128 * 128 * 128)   // voxels per sample = 2,097,152
#define B_SAMP 2
#define NLBL   6                   // label values 0..5 (bg + 5 blobs)
#define BPS    1024                // blocks per sample
#define TPB    256                 // 8 wave32 per block
#define VPT    8                   // voxels per thread; BPS*TPB*VPT == NVOX
#define ROWLEN 16                  // padded partial row (12 used: cnt[0..5], sum_p1[0..5])

typedef __attribute__((ext_vector_type(2))) float v2f;
typedef __attribute__((ext_vector_type(8))) float v8f;

__device__ __forceinline__ float waveRed32(float v) {
#pragma unroll
  for (int off = 16; off > 0; off >>= 1) v += __shfl_xor(v, off, 32);
  return v;
}

// Streaming pass: per block, compute cnt[L] and sum_p1[L] over its 2048 voxels,
// write one 16-float partial row to ws. Pure bandwidth-bound: 128-bit coalesced
// loads, v_exp_f32 sigmoid, predicated register accumulation, wave shuffle tree,
// ds_add_f32 across waves. No global atomics -> bitwise deterministic.
__global__ __launch_bounds__(TPB) void blob_partials_kernel(
    const float* __restrict__ x, const int* __restrict__ ml,
    float* __restrict__ ws) {
  const int sample = (int)blockIdx.x / BPS;
  const int blk    = (int)blockIdx.x % BPS;
  const int tid    = (int)threadIdx.x;

  const float* __restrict__ x0 = x + (size_t)(sample * 2 + 0) * NVOX;
  const float* __restrict__ x1 = x + (size_t)(sample * 2 + 1) * NVOX;
  const int*   __restrict__ lb = ml + (size_t)sample * NVOX;

  float cnt[NLBL], sm[NLBL];
#pragma unroll
  for (int i = 0; i < NLBL; ++i) { cnt[i] = 0.0f; sm[i] = 0.0f; }

  const int base = blk * (TPB * VPT);
#pragma unroll
  for (int pass = 0; pass < VPT / 4; ++pass) {
    const int v = base + pass * (TPB * 4) + tid * 4;
    const float4 a0 = *reinterpret_cast<const float4*>(x0 + v);
    const float4 a1 = *reinterpret_cast<const float4*>(x1 + v);
    const int4   lv = *reinterpret_cast<const int4*>(lb + v);

    float p[4];
    p[0] = 1.0f / (1.0f + __expf(a0.x - a1.x));   // softmax[1] = sigmoid(x1-x0)
    p[1] = 1.0f / (1.0f + __expf(a0.y - a1.y));
    p[2] = 1.0f / (1.0f + __expf(a0.z - a1.z));
    p[3] = 1.0f / (1.0f + __expf(a0.w - a1.w));
    int l[4] = {lv.x, lv.y, lv.z, lv.w};

#pragma unroll
    for (int e = 0; e < 4; ++e) {
#pragma unroll
      for (int L = 0; L < NLBL; ++L) {
        const bool m = (l[e] == L);
        cnt[L] += m ? 1.0f : 0.0f;
        sm[L]  += m ? p[e] : 0.0f;
      }
    }
  }

  __shared__ float sh[ROWLEN];
  if (tid < ROWLEN) sh[tid] = 0.0f;
  __syncthreads();

#pragma unroll
  for (int i = 0; i < NLBL; ++i) { cnt[i] = waveRed32(cnt[i]); sm[i] = waveRed32(sm[i]); }

  if ((tid & 31) == 0) {
#pragma unroll
    for (int i = 0; i < NLBL; ++i) {
      atomicAdd(&sh[i],        cnt[i]);   // ds_add_f32
      atomicAdd(&sh[NLBL + i], sm[i]);
    }
  }
  __syncthreads();

  if (tid < ROWLEN)
    ws[((size_t)(sample * BPS + blk)) * ROWLEN + tid] = sh[tid];  // pads 12..15 = 0
}

// Final reduction on the matrix pipe: C += ones(16x4) x B(4x16) sums 4 partial
// rows per v_wmma_f32_16x16x4_f32 (exact fp32). A==ones makes the result
// invariant to the K-slot mapping; only the N = lane&15 column mapping is used.
// Two independent accumulators (sample 0 / sample 1) keep the WMMA chain busy.
__global__ __launch_bounds__(32) void blob_finalize_kernel(
    const float* __restrict__ ws, float* __restrict__ out) {
  const int lane = (int)threadIdx.x;   // single wave32, EXEC all ones for WMMA
  const int col  = lane & 15;
  const int half = lane >> 4;

  v2f a; a[0] = 1.0f; a[1] = 1.0f;     // all-ones A (16x4 f32)
  v8f c0 = {}; v8f c1 = {};

  for (int chunk = 0; chunk < BPS / 4; ++chunk) {
    const float* r0 = ws + (size_t)(chunk * 4) * ROWLEN;          // sample 0
    const float* r1 = ws + (size_t)(BPS + chunk * 4) * ROWLEN;    // sample 1
    v2f b0, b1;
    b0[0] = r0[(half * 2 + 0) * ROWLEN + col];
    b0[1] = r0[(half * 2 + 1) * ROWLEN + col];
    b1[0] = r1[(half * 2 + 0) * ROWLEN + col];
    b1[1] = r1[(half * 2 + 1) * ROWLEN + col];
    c0 = __builtin_amdgcn_wmma_f32_16x16x4_f32(false, a, false, b0, (short)0, c0, false, false);
    c1 = __builtin_amdgcn_wmma_f32_16x16x4_f32(false, a, false, b1, (short)0, c1, false, false);
  }

  __shared__ float T[B_SAMP][16];
  if (lane < 16) { T[0][lane] = c0[0]; T[1][lane] = c1[0]; }  // D row 0 = column sums
  __syncthreads();

  if (lane == 0) {
    const double Nv = (double)NVOX;
    double total = 0.0;
    for (int b = 0; b < B_SAMP; ++b) {
      double cnt[NLBL], sm[NLBL];
      for (int i = 0; i < NLBL; ++i) { cnt[i] = (double)T[b][i]; sm[i] = (double)T[b][NLBL + i]; }

      double sp1 = 0.0, sg1 = 0.0, inter1 = 0.0;
      for (int L = 0; L < NLBL; ++L) sp1 += sm[L];
      for (int L = 1; L < NLBL; ++L) { sg1 += cnt[L]; inter1 += sm[L]; }
      const double inter0 = cnt[0] - sm[0];
      const double sp0 = Nv - sp1, sg0 = Nv - sg1;
      const double dc1 = 2.0 * inter1 / fmax(sp1 + sg1, 1e-8);
      const double dc0 = 2.0 * inter0 / fmax(sp0 + sg0, 1e-8);
      const double main_b = -0.5 * (dc0 + dc1);

      double blob_b = 0.0;
      for (int k = 1; k < NLBL; ++k) {
        const double other = sg1 - cnt[k];               // fg voxels of other blobs
        const double sp1k  = sm[0] + sm[k] + 0.5 * other;
        const double i1k   = sm[k];
        const double i0k   = (cnt[0] - sm[0]) + 0.5 * other;
        const double sp0k  = Nv - sp1k;
        const double d1k = 2.0 * i1k / fmax(sp1k + cnt[k],        1e-8);
        const double d0k = 2.0 * i0k / fmax(sp0k + (Nv - cnt[k]), 1e-8);
        blob_b += -0.5 * (d0k + d1k);
      }
      blob_b /= 5.0;  // mean over K blobs

      // ALPHA*DICE_W = 1.0 (main), BETA*DICE_W = 0.5 (blob), per sample
      total += 1.0 * main_b + 0.5 * blob_b;
    }
    out[0] = (float)(total / (double)B_SAMP);
  }
}

extern "C" void kernel_launch(void* const* d_in, const int* in_sizes, int n_in,
                              void* d_out, int out_size, void* d_ws, size_t ws_size,
                              hipStream_t stream) {
  (void)in_sizes; (void)n_in; (void)out_size; (void)ws_size;
  const float* x  = (const float*)d_in[0];   // [2,2,128^3] f32
  const int*   ml = (const int*)d_in[2];     // [2,1,128^3] i32 (y == ml>0 is redundant)
  float* ws  = (float*)d_ws;                 // [2*BPS][16] partials = 128 KB
  float* out = (float*)d_out;                // scalar

  blob_partials_kernel<<<B_SAMP * BPS, TPB, 0, stream>>>(x, ml, ws);
  blob_finalize_kernel<<<1, 32, 0, stream>>>(ws, out);
}